// GATLayer_6416681140653
// MI455X (gfx1250) — compile-verified
//
#include <hip/hip_runtime.h>
#include <math.h>

#define DIM 64

typedef __attribute__((ext_vector_type(2))) float v2f;
typedef __attribute__((ext_vector_type(8))) float v8f;

// Order-preserving float <-> uint encoding so we can use atomicMax(u32) for
// a float segment-max (handles negatives from leaky_relu).
__device__ __forceinline__ unsigned enc_f32(float f) {
  unsigned u = __float_as_uint(f);
  return (u & 0x80000000u) ? ~u : (u | 0x80000000u);
}
__device__ __forceinline__ float dec_f32(unsigned u) {
  u = (u & 0x80000000u) ? (u ^ 0x80000000u) : ~u;
  return __uint_as_float(u);
}

// Zero the accumulator output and per-node scratch (runs every call; the
// harness does not re-zero between replays).
__global__ void k_init(float* __restrict__ out, float* __restrict__ denom,
                       unsigned* __restrict__ emax, int N) {
  int t = blockIdx.x * blockDim.x + threadIdx.x;
  if (t < N * DIM) out[t] = 0.f;
  if (t < N) { denom[t] = 0.f; emax[t] = 0u; }  // 0u == enc(most-negative)
}

// s_src / s_dst for 16 nodes per wave via V_WMMA_F32_16X16X4_F32.
// A (16x4 f32): lane L -> row M=L&15, VGPRs hold K = (L>>4)*2 + {0,1}.
// B (4x16 f32): lane L -> col N=L&15, VGPRs hold K = (L>>4)*2 + {0,1}.
//   col 0 = a_src chunk, col 1 = a_dst chunk, cols 2..15 = 0 (branchless
//   cndmask on loaded values — keeps the k-loop free of EXEC manipulation).
// D (16x16 f32): lane L holds col N=L&15; VGPR r = row r + 8*(L>>4).
__global__ void k_scores_wmma(const float* __restrict__ feat,
                              const float* __restrict__ attn,
                              float* __restrict__ s_src,
                              float* __restrict__ s_dst, int N) {
  int lane  = threadIdx.x & 31;
  int wave  = (blockIdx.x * blockDim.x + threadIdx.x) >> 5;
  int base  = wave * 16;
  int m     = lane & 15;   // A row == B/D column for this lane
  int khalf = lane >> 4;

  int row = base + m;
  if (row >= N) row = N - 1;          // clamp: keeps all lanes active (EXEC all-1s)
  const float* frow = feat + (long)row * DIM;

  // Per-lane B base: column 0 reads a_src, column 1 reads a_dst; all other
  // lanes still load (attn is 512 B, cache-resident) but select 0.
  const float* brow = attn + ((m == 1) ? DIM : 0);
  bool bvalid = (m < 2);

  v8f c = {};
#pragma unroll
  for (int kk = 0; kk < 16; ++kk) {
    int k0 = kk * 4 + khalf * 2;
    v2f a;
    a.x = frow[k0];
    a.y = frow[k0 + 1];
    float bx = brow[k0];
    float by = brow[k0 + 1];
    v2f b;
    b.x = bvalid ? bx : 0.f;
    b.y = bvalid ? by : 0.f;
    c = __builtin_amdgcn_wmma_f32_16x16x4_f32(
        /*neg_a=*/false, a, /*neg_b=*/false, b,
        /*c_mod=*/(short)0, c, /*reuse_a=*/false, /*reuse_b=*/false);
  }

  if (m < 2) {
    float* outp = (m == 0) ? s_src : s_dst;
    int node0 = base + khalf * 8;
#pragma unroll
    for (int r = 0; r < 8; ++r) {
      int node = node0 + r;
      if (node < N) outp[node] = c[r];
    }
  }
}

// Pass 1 over edges: e = leaky_relu(s_src[src]+s_dst[dst]); segment-max(dst).
__global__ void k_edge_max(const int* __restrict__ src, const int* __restrict__ dst,
                           const float* __restrict__ s_src,
                           const float* __restrict__ s_dst,
                           float* __restrict__ e_out, unsigned* __restrict__ emax,
                           int E) {
  int i = blockIdx.x * blockDim.x + threadIdx.x;
  if (i >= E) return;
  int s = src[i], d = dst[i];
  float e = s_src[s] + s_dst[d];
  e = (e > 0.f) ? e : 0.01f * e;
  e_out[i] = e;
  atomicMax(&emax[d], enc_f32(e));
}

// Pass 2 over edges: ee = exp(e - emax[dst]); segment-sum(dst) -> denom.
__global__ void k_edge_exp(const int* __restrict__ dst,
                           float* __restrict__ ee,
                           const unsigned* __restrict__ emax,
                           float* __restrict__ denom, int E) {
  int i = blockIdx.x * blockDim.x + threadIdx.x;
  if (i >= E) return;
  int d = dst[i];
  float ex = __expf(ee[i] - dec_f32(emax[d]));
  ee[i] = ex;
  atomicAdd(&denom[d], ex);
}

// Pass 3: one wave per edge; 32 lanes x float2 cover the 64-wide feature row.
// Features/out stay L2-resident (25.6 MB each vs 192 MB L2); prefetch the
// next edge's feature row (global_prefetch_b8).
__global__ void k_aggregate(const int* __restrict__ src, const int* __restrict__ dst,
                            const float* __restrict__ feat,
                            const float* __restrict__ ee,
                            const float* __restrict__ denom,
                            float* __restrict__ out, int E) {
  int lane   = threadIdx.x & 31;
  int wid    = (blockIdx.x * blockDim.x + threadIdx.x) >> 5;
  int nwaves = (gridDim.x * blockDim.x) >> 5;
  const float2* feat2 = (const float2*)feat;
  for (int e = wid; e < E; e += nwaves) {
    int s = src[e], d = dst[e];
    int e2 = e + nwaves;
    if (e2 < E) __builtin_prefetch(&feat[(long)src[e2] * DIM], 0, 0);
    float alpha = ee[e] / denom[d];
    float2 f = feat2[(long)s * (DIM / 2) + lane];
    atomicAdd(&out[(long)d * DIM + lane * 2 + 0], alpha * f.x);
    atomicAdd(&out[(long)d * DIM + lane * 2 + 1], alpha * f.y);
  }
}

// Final: out = elu(h) in place.
__global__ void k_elu(float* __restrict__ out, int total) {
  int t = blockIdx.x * blockDim.x + threadIdx.x;
  if (t < total) {
    float h = out[t];
    out[t] = (h > 0.f) ? h : (__expf(h) - 1.f);
  }
}

extern "C" void kernel_launch(void* const* d_in, const int* in_sizes, int n_in,
                              void* d_out, int out_size, void* d_ws, size_t ws_size,
                              hipStream_t stream) {
  const float* feat = (const float*)d_in[0];
  const float* attn = (const float*)d_in[1];
  const int*   src  = (const int*)d_in[2];
  const int*   dst  = (const int*)d_in[3];
  int N = in_sizes[0] / DIM;
  int E = in_sizes[2];
  float* out = (float*)d_out;

  // Workspace layout (floats): s_src[N] | s_dst[N] | denom[N] | emax[N](u32) | ee[E]
  float*    ws    = (float*)d_ws;
  float*    s_src = ws;
  float*    s_dst = ws + (size_t)N;
  float*    denom = ws + 2 * (size_t)N;
  unsigned* emax  = (unsigned*)(ws + 3 * (size_t)N);
  float*    ee    = ws + 4 * (size_t)N;

  const int tpb = 256;  // 8 waves/block on wave32

  k_init<<<(N * DIM + tpb - 1) / tpb, tpb, 0, stream>>>(out, denom, emax, N);

  int waves_needed = (N + 15) / 16;
  int blocks_w = (waves_needed + 7) / 8;  // 8 waves per 256-thread block
  k_scores_wmma<<<blocks_w, tpb, 0, stream>>>(feat, attn, s_src, s_dst, N);

  k_edge_max<<<(E + tpb - 1) / tpb, tpb, 0, stream>>>(src, dst, s_src, s_dst, ee, emax, E);
  k_edge_exp<<<(E + tpb - 1) / tpb, tpb, 0, stream>>>(dst, ee, emax, denom, E);
  k_aggregate<<<4096, tpb, 0, stream>>>(src, dst, feat, ee, denom, out, E);
  k_elu<<<(N * DIM + tpb - 1) / tpb, tpb, 0, stream>>>(out, N * DIM);
}